// GraphAttentionLayer_46316927320306
// MI455X (gfx1250) — compile-verified
//
#include <hip/hip_runtime.h>
#include <hip/hip_bf16.h>

#define N_NODES 8192
#define E_EDGES 262144
#define IN_DIM 256
#define OUT_DIM 256
#define NEG_SLOPE 0.2f

typedef float v2f __attribute__((ext_vector_type(2)));
typedef float v8f __attribute__((ext_vector_type(8)));

// ---------------------------------------------------------------------------
// Kernel 0: zero accumulator regions of the workspace.
// ---------------------------------------------------------------------------
__global__ __launch_bounds__(256) void gat_zero_ws(float* __restrict__ p, int n) {
    int i = blockIdx.x * blockDim.x + threadIdx.x;
    int stride = gridDim.x * blockDim.x;
    for (; i < n; i += stride) p[i] = 0.0f;
}

// ---------------------------------------------------------------------------
// Kernel 1: h = x @ W (8192x256 @ 256x256 fp32) via V_WMMA_F32_16X16X4_F32.
//
// Block = 256 threads (8 waves). Each block owns one N-panel (tn) and 8
// consecutive M-tiles. The 256x16 fp32 W panel (16 KB) is staged into LDS
// with gfx1250 async global->LDS copies (ASYNCcnt), then each wave runs the
// K loop with A from global (b64/lane) and B from LDS (ds_load).
//
// A (16x4 f32): lane l holds A[l%16][2*(l/16)+{0,1}]
// B (4x16 f32): lane l holds B[2*(l/16)+{0,1}][l%16]
// C/D (16x16 f32): VGPR r on lane l is D[r + 8*(l/16)][l%16]
// ---------------------------------------------------------------------------
__global__ __launch_bounds__(256) void gat_gemm_h(const float* __restrict__ X,
                                                  const float* __restrict__ W,
                                                  float* __restrict__ H) {
    __shared__ float Bs[IN_DIM * 16];                // 16 KB panel: Bs[k*16 + n]

    const int lane = threadIdx.x & 31;
    const int wave = threadIdx.x >> 5;               // 0..7
    const int tn   = blockIdx.x & 15;                // N-panel id (0..15)
    const int tg   = blockIdx.x >> 4;                // 0..63
    const int tm   = tg * 8 + wave;                  // M-tile id (0..511)

    // ---- stage W panel into LDS with async global->LDS b128 copies --------
    // panel float4 f (0..1023): k = f/4, ngrp = f%4
    {
        const int t = threadIdx.x;                   // 0..255
#pragma unroll
        for (int r = 0; r < 4; ++r) {
            const int f    = r * 256 + t;
            const int k    = f >> 2;
            const int ngrp = f & 3;
            const float* gsrc = W + (size_t)k * OUT_DIM + tn * 16 + ngrp * 4;
            const unsigned lds_off = (unsigned)(size_t)(&Bs[f * 4]);
            asm volatile("global_load_async_to_lds_b128 %0, %1, off"
                         :
                         : "v"(lds_off), "v"(gsrc)
                         : "memory");
        }
        asm volatile("s_wait_asynccnt 0x0" ::: "memory");
    }
    __syncthreads();

    // ---- WMMA K loop ------------------------------------------------------
    const int mrow  = tm * 16 + (lane & 15);
    const int khalf = (lane >> 4) * 2;               // 0 or 2
    const int nlane = lane & 15;
    const int ncol  = tn * 16 + nlane;

    const float* __restrict__ xrow = X + (size_t)mrow * IN_DIM;

    v8f acc = {};
#pragma unroll 4
    for (int kk = 0; kk < IN_DIM; kk += 4) {
        __builtin_prefetch(xrow + kk + 64, 0, 3);    // global_prefetch_b8
        v2f a, b;
        a.x = xrow[kk + khalf];
        a.y = xrow[kk + khalf + 1];
        b.x = Bs[(kk + khalf) * 16 + nlane];
        b.y = Bs[(kk + khalf + 1) * 16 + nlane];
        // (neg_a, A, neg_b, B, c_mod, C, reuse_a, reuse_b)
        acc = __builtin_amdgcn_wmma_f32_16x16x4_f32(false, a, false, b,
                                                    (short)0, acc, false, false);
    }

    const int rbase = (lane >> 4) * 8;
#pragma unroll
    for (int r = 0; r < 8; ++r) {
        H[(size_t)(tm * 16 + rbase + r) * OUT_DIM + ncol] = acc[r];
    }
}

// ---------------------------------------------------------------------------
// Kernel 2: per-node scores s_src[i] = h[i].a_src, s_tgt[i] = h[i].a_tgt.
// One wave32 per node, shuffle reduction.
// ---------------------------------------------------------------------------
__global__ __launch_bounds__(256) void gat_node_scores(const float* __restrict__ H,
                                                       const float* __restrict__ a,
                                                       float* __restrict__ s_src,
                                                       float* __restrict__ s_tgt) {
    const int node = blockIdx.x * 8 + (threadIdx.x >> 5);
    const int lane = threadIdx.x & 31;
    const float* __restrict__ hr = H + (size_t)node * OUT_DIM;
    float ss = 0.0f, st = 0.0f;
#pragma unroll
    for (int j = lane; j < OUT_DIM; j += 32) {
        const float hv = hr[j];
        ss += hv * a[j];
        st += hv * a[OUT_DIM + j];
    }
#pragma unroll
    for (int off = 16; off > 0; off >>= 1) {
        ss += __shfl_down(ss, off, 32);
        st += __shfl_down(st, off, 32);
    }
    if (lane == 0) { s_src[node] = ss; s_tgt[node] = st; }
}

// ---------------------------------------------------------------------------
// Kernel 3: column sums of h (Hsum[d] = sum_i h[i][d]).
// ---------------------------------------------------------------------------
__global__ __launch_bounds__(256) void gat_col_sum(const float* __restrict__ H,
                                                   float* __restrict__ Hsum) {
    const int d  = threadIdx.x;
    const int r0 = blockIdx.x * 256;
    float s = 0.0f;
    for (int i = 0; i < 256; ++i) s += H[(size_t)(r0 + i) * OUT_DIM + d];
    atomicAdd(&Hsum[d], s);
}

// ---------------------------------------------------------------------------
// Kernel 4: per-edge weight + scatter (analytic softmax decomposition).
//   w = exp(leaky_relu(s_src[src]+s_tgt[tgt])) - 1
//   denomExtra[src] += w ;  num[src][:] += w * h[tgt][:]
// ---------------------------------------------------------------------------
__global__ __launch_bounds__(256) void gat_edge_scatter(const long long* __restrict__ ei,
                                                        const float* __restrict__ H,
                                                        const float* __restrict__ s_src,
                                                        const float* __restrict__ s_tgt,
                                                        float* __restrict__ num,
                                                        float* __restrict__ denomExtra) {
    const int e    = blockIdx.x * 8 + (threadIdx.x >> 5);
    const int lane = threadIdx.x & 31;
    const int src  = (int)ei[e];
    const int tgt  = (int)ei[E_EDGES + e];

    float sc = s_src[src] + s_tgt[tgt];
    sc = (sc >= 0.0f) ? sc : NEG_SLOPE * sc;
    const float w = __expf(sc) - 1.0f;

    if (lane == 0) atomicAdd(&denomExtra[src], w);

    const float* __restrict__ ht = H + (size_t)tgt * OUT_DIM;
    float* __restrict__ nr = num + (size_t)src * OUT_DIM;
#pragma unroll
    for (int j = lane; j < OUT_DIM; j += 32) {
        atomicAdd(&nr[j], w * ht[j]);
    }
}

// ---------------------------------------------------------------------------
// Kernel 5: finalize  out[i][d] = (Hsum[d] + num[i][d]) / (N + denomExtra[i])
// ---------------------------------------------------------------------------
__global__ __launch_bounds__(256) void gat_finalize(const float* __restrict__ num,
                                                    const float* __restrict__ denomExtra,
                                                    const float* __restrict__ Hsum,
                                                    float* __restrict__ out) {
    const int i = blockIdx.x;
    const int d = threadIdx.x;
    const float inv = 1.0f / ((float)N_NODES + denomExtra[i]);
    out[(size_t)i * OUT_DIM + d] = (Hsum[d] + num[(size_t)i * OUT_DIM + d]) * inv;
}

extern "C" void kernel_launch(void* const* d_in, const int* in_sizes, int n_in,
                              void* d_out, int out_size, void* d_ws, size_t ws_size,
                              hipStream_t stream) {
    const float*     x  = (const float*)d_in[0];
    const long long* ei = (const long long*)d_in[1];   // int64 (2, E)
    const float*     W  = (const float*)d_in[2];
    const float*     a  = (const float*)d_in[3];
    float* out = (float*)d_out;

    // workspace layout (floats):
    float* ws         = (float*)d_ws;
    float* H          = ws;                                  // N*256
    float* num        = H   + (size_t)N_NODES * OUT_DIM;     // N*256
    float* s_src      = num + (size_t)N_NODES * OUT_DIM;     // N
    float* s_tgt      = s_src + N_NODES;                     // N
    float* denomExtra = s_tgt + N_NODES;                     // N
    float* Hsum       = denomExtra + N_NODES;                // 256

    gat_zero_ws<<<2048, 256, 0, stream>>>(num, N_NODES * OUT_DIM);
    gat_zero_ws<<<64, 256, 0, stream>>>(denomExtra, N_NODES + 256);

    // h = x @ W : 16 N-panels x 64 M-groups, 8 waves (M-tiles) per block
    gat_gemm_h<<<16 * 64, 256, 0, stream>>>(x, W, H);

    gat_node_scores<<<N_NODES / 8, 256, 0, stream>>>(H, a, s_src, s_tgt);
    gat_col_sum<<<N_NODES / 256, 256, 0, stream>>>(H, Hsum);
    gat_edge_scatter<<<E_EDGES / 8, 256, 0, stream>>>(ei, H, s_src, s_tgt, num, denomExtra);
    gat_finalize<<<N_NODES, 256, 0, stream>>>(num, denomExtra, Hsum, out);
}